// SSM_46102178955725
// MI455X (gfx1250) — compile-verified
//
#include <hip/hip_runtime.h>

// ---------------------------------------------------------------------------
// Linear SSM  h_t = h_{t-1} A^T + x_t B^T ; y_t = h_t C^T
// BATCH=16, SEQ=4096, D_IN=N_STATE=D_OUT=256, fp32.
//
// MI455X plan: chunked parallel scan with fp32 WMMA (V_WMMA_F32_16X16X4_F32),
// batch=16 == WMMA M dim. Per-chunk state h[16][256] lives in LDS, now
// PING-PONGED (one barrier per step instead of two). The constant B-operand
// fragments of A^T are cached in 64 v2f registers per wave, and the u_t tile
// is register double-buffered (loads for t+1 overlap step t's WMMA chain,
// global_prefetch_b8 warms t+2), so the serial critical path per step is
// ds_load_b64 + chained wmma + 1 barrier.
// U = X B^T is staged inside d_out (overlap proof in k_scan_y comments).
// Workspace use: 2*256KB (A^64 ping-pong) + 1MB (E) + 1MB (S_c) = 2.5 MB.
// ---------------------------------------------------------------------------

typedef float v2f __attribute__((ext_vector_type(2)));
typedef float v8f __attribute__((ext_vector_type(8)));

#define BATCH 16
#define SEQ   4096
#define D     256          // D_IN == N_STATE == D_OUT
#define LCH   64           // chunk length (2*LCH + SEQ/LCH ~ minimal serial depth)
#define NC    (SEQ / LCH)  // 64 chunks
#define KT    (D / 4)      // 64 k-steps of the 16x16x4 fp32 WMMA
#define NT    (D / 16)     // 16 column tiles
#define PAD   260          // LDS row pitch in floats; (4*ln + 2*hi + {0,1}) hits
                           // all 64 banks exactly once for the b64 A-frag loads

__device__ __forceinline__ v8f wmma4(v2f a, v2f b, v8f c) {
  // D(16x16,f32) += A(16x4,f32) * B(4x16,f32)
  return __builtin_amdgcn_wmma_f32_16x16x4_f32(
      /*neg_a=*/false, a, /*neg_b=*/false, b,
      /*c_mod=*/(short)0, c, /*reuse_a=*/false, /*reuse_b=*/false);
}

// ---------------------------------------------------------------------------
// U = X @ B^T : [BATCH*SEQ, 256] x [256, 256]^T. One wave = 16 rows x all 256
// cols; A-frags for the 16-row strip cached in 64 v2f regs (loaded once).
// Bandwidth-bound (X streamed once, Bm L1/L2-hot, U stays L2-resident for the
// scan passes -> default RT temporal hints on purpose).
// ---------------------------------------------------------------------------
__global__ void __launch_bounds__(256) k_gemm_u(const float* __restrict__ X,
                                                const float* __restrict__ Bm,
                                                float* __restrict__ U) {
  const int wave = (int)((blockIdx.x * blockDim.x + threadIdx.x) >> 5);
  const int lane = (int)(threadIdx.x & 31);
  const int ln = lane & 15, hi = lane >> 4;
  const int r0 = wave * 16;                       // 16 rows of X / U

  v2f a[KT];                                      // A-frags: row ln, k=4kk+2hi+{0,1}
  const float* xrow = X + (size_t)(r0 + ln) * D + 2 * hi;
#pragma unroll
  for (int kk = 0; kk < KT; ++kk) a[kk] = *(const v2f*)(xrow + 4 * kk);

  for (int nt = 0; nt < NT; ++nt) {
    // B-operand[k][n] = Bm[n][k] -> consecutive k in row n of Bm (b64 loads)
    const float* brow = Bm + (size_t)(nt * 16 + ln) * D + 2 * hi;
    v8f acc = {};
#pragma unroll
    for (int kk = 0; kk < KT; ++kk) {
      v2f b = *(const v2f*)(brow + 4 * kk);
      acc = wmma4(a[kk], b, acc);
    }
    float* urow = U + (size_t)r0 * D + nt * 16 + ln;
#pragma unroll
    for (int r = 0; r < 8; ++r) urow[(size_t)(r + 8 * hi) * D] = acc[r];
  }
}

// ---------------------------------------------------------------------------
// Dst = S @ S (256x256 row-major). Repeated squaring builds A^64 in 6 launches.
// 64 waves x 4 tiles = 256 output tiles.
// ---------------------------------------------------------------------------
__global__ void __launch_bounds__(128) k_apow(const float* __restrict__ S,
                                              float* __restrict__ Dst) {
  const int wid  = (int)((blockIdx.x * blockDim.x + threadIdx.x) >> 5);  // 0..63
  const int lane = (int)(threadIdx.x & 31);
  const int ln = lane & 15, hi = lane >> 4;
  for (int t = 0; t < 4; ++t) {
    const int tile = wid * 4 + t;                 // 0..255
    const int i0 = (tile >> 4) * 16, j0 = (tile & 15) * 16;
    const float* arow = S + (size_t)(i0 + ln) * D + 2 * hi;
    v8f acc = {};
#pragma unroll
    for (int kk = 0; kk < KT; ++kk) {
      v2f a = *(const v2f*)(arow + 4 * kk);
      const int k = 4 * kk + 2 * hi;
      v2f b;                                      // B[k][j] = S[k*D+j] (column pair)
      b.x = S[(size_t)k * D + j0 + ln];
      b.y = S[(size_t)(k + 1) * D + j0 + ln];
      acc = wmma4(a, b, acc);
    }
    float* drow = Dst + (size_t)i0 * D + j0 + ln;
#pragma unroll
    for (int r = 0; r < 8; ++r) drow[(size_t)(r + 8 * hi) * D] = acc[r];
  }
}

// ---------------------------------------------------------------------------
// Pass 1: per chunk, zero-init local scan; emit only the chunk-end state E_c.
// One workgroup per chunk; wave w owns columns [16w, 16w+16). h ping-pongs
// between two LDS buffers -> exactly one barrier per time step.
// ---------------------------------------------------------------------------
__global__ void __launch_bounds__(512) k_chunk_end(const float* __restrict__ U,
                                                   const float* __restrict__ Am,
                                                   float* __restrict__ E) {
  __shared__ float h[2][16 * PAD];
  const int c = (int)blockIdx.x;
  const int w = (int)(threadIdx.x >> 5);
  const int lane = (int)(threadIdx.x & 31);
  const int ln = lane & 15, hi = lane >> 4;
  const int n0 = w * 16;

  // Cache B-frags of M = A^T for this column tile: B[k][n] = Am[n][k].
  v2f m[KT];
  const float* amrow = Am + (size_t)(n0 + ln) * D + 2 * hi;
#pragma unroll
  for (int kk = 0; kk < KT; ++kk) m[kk] = *(const v2f*)(amrow + 4 * kk);

#pragma unroll
  for (int r = 0; r < 8; ++r) h[0][(r + 8 * hi) * PAD + n0 + ln] = 0.0f;

  // Software-pipelined u-tile: preload t=0.
  v8f unext;
#pragma unroll
  for (int r = 0; r < 8; ++r)
    unext[r] = U[((size_t)(r + 8 * hi) * SEQ + (size_t)c * LCH) * D + n0 + ln];
  __syncthreads();

  int p = 0;
  v8f acc = {};
  for (int t = 0; t < LCH; ++t) {
    const int ta = c * LCH + t;
    acc = unext;                                   // u_t tile = C-operand init
    if (t + 1 < LCH) {                             // overlap t+1 loads with WMMAs
#pragma unroll
      for (int r = 0; r < 8; ++r)
        unext[r] = U[((size_t)(r + 8 * hi) * SEQ + ta + 1) * D + n0 + ln];
      if (t + 2 < LCH) {                           // global_prefetch_b8 for t+2
#pragma unroll
        for (int r = 0; r < 8; ++r)
          __builtin_prefetch(&U[((size_t)(r + 8 * hi) * SEQ + ta + 2) * D + n0 + ln], 0, 1);
      }
    }
#pragma unroll
    for (int kk = 0; kk < KT; ++kk) {              // critical path: ds + wmma
      v2f a = *(const v2f*)&h[p][ln * PAD + 4 * kk + 2 * hi];
      acc = wmma4(a, m[kk], acc);
    }
#pragma unroll
    for (int r = 0; r < 8; ++r) h[1 - p][(r + 8 * hi) * PAD + n0 + ln] = acc[r];
    __syncthreads();                               // publish h_t; one barrier/step
    p ^= 1;
  }
  float* erow = E + (size_t)c * (16 * D) + n0 + ln;
#pragma unroll
  for (int r = 0; r < 8; ++r) erow[(size_t)(r + 8 * hi) * D] = acc[r];
}

// ---------------------------------------------------------------------------
// Pass 2 (single workgroup): S_0 = 0 ; S_{c+1} = S_c (A^T)^L + E_c.
// AL is row-major A^64, so B[k][n] = AL[n][k] (b64 loads, cached in regs).
// ---------------------------------------------------------------------------
__global__ void __launch_bounds__(512) k_boundary(const float* __restrict__ E,
                                                  const float* __restrict__ AL,
                                                  float* __restrict__ Sc) {
  __shared__ float h[2][16 * PAD];
  const int w = (int)(threadIdx.x >> 5);
  const int lane = (int)(threadIdx.x & 31);
  const int ln = lane & 15, hi = lane >> 4;
  const int n0 = w * 16;

  v2f m[KT];
  const float* alrow = AL + (size_t)(n0 + ln) * D + 2 * hi;
#pragma unroll
  for (int kk = 0; kk < KT; ++kk) m[kk] = *(const v2f*)(alrow + 4 * kk);

  v8f cur = {};
#pragma unroll
  for (int r = 0; r < 8; ++r) h[0][(r + 8 * hi) * PAD + n0 + ln] = 0.0f;
  __syncthreads();

  int p = 0;
  for (int c = 0; c < NC; ++c) {
    float* srow = Sc + (size_t)c * (16 * D) + n0 + ln;
#pragma unroll
    for (int r = 0; r < 8; ++r) srow[(size_t)(r + 8 * hi) * D] = cur[r];

    v8f acc;
    const float* erow = E + (size_t)c * (16 * D) + n0 + ln;
#pragma unroll
    for (int r = 0; r < 8; ++r) acc[r] = erow[(size_t)(r + 8 * hi) * D];
#pragma unroll
    for (int kk = 0; kk < KT; ++kk) {
      v2f a = *(const v2f*)&h[p][ln * PAD + 4 * kk + 2 * hi];
      acc = wmma4(a, m[kk], acc);
    }
#pragma unroll
    for (int r = 0; r < 8; ++r) h[1 - p][(r + 8 * hi) * PAD + n0 + ln] = acc[r];
    __syncthreads();
    p ^= 1;
    cur = acc;
  }
}

// ---------------------------------------------------------------------------
// Pass 3: re-scan each chunk from S_c and fuse y_t = h_t C^T.
// UY is d_out, aliased: holds U on entry, y on exit. Safety with one barrier
// per step: every wave's s_wait_loadcnt for its u_t (and in-flight u_{t+1},
// address ta+1 != ta) precedes barrier(t); y_t stores to ta follow barrier(t);
// chunks touch disjoint t-ranges; written y is never re-read.
// M-frags register-cached (critical path); C-frags streamed (L1-hot, off the
// critical path) to keep VGPR pressure moderate.
// ---------------------------------------------------------------------------
__global__ void __launch_bounds__(512) k_scan_y(const float* __restrict__ Sc,
                                                const float* __restrict__ Am,
                                                const float* __restrict__ Cm,
                                                float* __restrict__ UY) {
  __shared__ float h[2][16 * PAD];
  const int c = (int)blockIdx.x;
  const int w = (int)(threadIdx.x >> 5);
  const int lane = (int)(threadIdx.x & 31);
  const int ln = lane & 15, hi = lane >> 4;
  const int n0 = w * 16;

  v2f m[KT];
  const float* amrow = Am + (size_t)(n0 + ln) * D + 2 * hi;
#pragma unroll
  for (int kk = 0; kk < KT; ++kk) m[kk] = *(const v2f*)(amrow + 4 * kk);
  const float* crow = Cm + (size_t)(n0 + ln) * D + 2 * hi;  // B[k][n]=Cm[n][k]

  {
    const float* srow = Sc + (size_t)c * (16 * D) + n0 + ln;
#pragma unroll
    for (int r = 0; r < 8; ++r)
      h[0][(r + 8 * hi) * PAD + n0 + ln] = srow[(size_t)(r + 8 * hi) * D];
  }
  v8f unext;
#pragma unroll
  for (int r = 0; r < 8; ++r)
    unext[r] = UY[((size_t)(r + 8 * hi) * SEQ + (size_t)c * LCH) * D + n0 + ln];
  __syncthreads();

  int p = 0;
  for (int t = 0; t < LCH; ++t) {
    const int ta = c * LCH + t;
    v8f acc = unext;                               // u_t
    if (t + 1 < LCH) {                             // overlap t+1 loads with WMMAs
#pragma unroll
      for (int r = 0; r < 8; ++r)
        unext[r] = UY[((size_t)(r + 8 * hi) * SEQ + ta + 1) * D + n0 + ln];
      if (t + 2 < LCH) {
#pragma unroll
        for (int r = 0; r < 8; ++r)
          __builtin_prefetch(&UY[((size_t)(r + 8 * hi) * SEQ + ta + 2) * D + n0 + ln], 0, 1);
      }
    }
#pragma unroll
    for (int kk = 0; kk < KT; ++kk) {              // h_t = h_{t-1} A^T + u_t
      v2f a = *(const v2f*)&h[p][ln * PAD + 4 * kk + 2 * hi];
      acc = wmma4(a, m[kk], acc);
    }
#pragma unroll
    for (int r = 0; r < 8; ++r) h[1 - p][(r + 8 * hi) * PAD + n0 + ln] = acc[r];
    __syncthreads();                               // publish h_t (single barrier)

    v8f y = {};
#pragma unroll
    for (int kk = 0; kk < KT; ++kk) {              // y_t = h_t C^T (reads h[1-p])
      v2f a = *(const v2f*)&h[1 - p][ln * PAD + 4 * kk + 2 * hi];
      v2f b = *(const v2f*)(crow + 4 * kk);
      y = wmma4(a, b, y);
    }
#pragma unroll
    for (int r = 0; r < 8; ++r)
      UY[((size_t)(r + 8 * hi) * SEQ + ta) * D + n0 + ln] = y[r];
    p ^= 1;
  }
}

// ---------------------------------------------------------------------------
extern "C" void kernel_launch(void* const* d_in, const int* in_sizes, int n_in,
                              void* d_out, int out_size, void* d_ws, size_t ws_size,
                              hipStream_t stream) {
  (void)in_sizes; (void)n_in; (void)out_size; (void)ws_size;
  const float* X  = (const float*)d_in[0];   // [16,4096,256]
  const float* Am = (const float*)d_in[1];   // [256,256]
  const float* Bm = (const float*)d_in[2];   // [256,256]
  const float* Cm = (const float*)d_in[3];   // [256,256]
  float* Y = (float*)d_out;                  // staged as U, finally y

  float* P0 = (float*)d_ws;                  // 256 KB
  float* P1 = P0 + D * D;                    // 256 KB
  float* E  = P1 + D * D;                    // 1 MB
  float* Sc = E + (size_t)NC * 16 * D;       // 1 MB  (total ws use: 2.5 MB)

  // Pass 0: U = X B^T  (4096 row-tiles, 8 waves/block)
  k_gemm_u<<<(BATCH * SEQ / 16) / 8, 256, 0, stream>>>(X, Bm, Y);

  // A^64 via repeated squaring (6 tiny GEMMs, ping-pong): result in P1.
  k_apow<<<16, 128, 0, stream>>>(Am, P0);
  k_apow<<<16, 128, 0, stream>>>(P0, P1);
  k_apow<<<16, 128, 0, stream>>>(P1, P0);
  k_apow<<<16, 128, 0, stream>>>(P0, P1);
  k_apow<<<16, 128, 0, stream>>>(P1, P0);
  k_apow<<<16, 128, 0, stream>>>(P0, P1);

  // Pass 1: per-chunk local end states (parallel over 64 chunks).
  k_chunk_end<<<NC, 512, 0, stream>>>(Y, Am, E);
  // Pass 2: serial chunk-boundary recurrence (single workgroup, 64 steps).
  k_boundary<<<1, 512, 0, stream>>>(E, P1, Sc);
  // Pass 3: re-scan each chunk from S_c, fused y = h C^T, y -> d_out.
  k_scan_y<<<NC, 512, 0, stream>>>(Sc, Am, Cm, Y);
}